// GraphGNN_80376017977456
// MI455X (gfx1250) — compile-verified
//
#include <hip/hip_runtime.h>

typedef __attribute__((ext_vector_type(2))) float v2f;
typedef __attribute__((ext_vector_type(8))) float v8f;

#define FOUT 20
#define NGRAPHS 512

// ---------------------------------------------------------------------------
// zero fill
// ---------------------------------------------------------------------------
__global__ void zero_f32(float* __restrict__ p, int n) {
    int i = blockIdx.x * blockDim.x + threadIdx.x;
    if (i < n) p[i] = 0.0f;
}

// ---------------------------------------------------------------------------
// edge scatter: agg[dst, :] += x[src, :] * ew[e]   (one thread per edge)
// Node feature arrays are <=8MB -> resident in the 192MB L2, so the random
// gather/scatter (atomic) traffic all lands in L2; the streamed edge list
// (~38MB/layer) is the only HBM-side traffic and is itself L2-cacheable
// across the three layers.
// ---------------------------------------------------------------------------
template<int F>
__global__ void edge_scatter(const int* __restrict__ src, const int* __restrict__ dst,
                             const float* __restrict__ ew, const float* __restrict__ x,
                             float* __restrict__ agg, int E) {
    int e = blockIdx.x * blockDim.x + threadIdx.x;
    if (e >= E) return;
    const int   s = src[e];
    const int   d = dst[e];
    const float w = ew[e];
    const v2f* __restrict__ xr = (const v2f*)(x + (long long)s * F);
    float* __restrict__ ar = agg + (long long)d * F;
    v2f row[F / 2];
    #pragma unroll
    for (int i = 0; i < F / 2; ++i) row[i] = xr[i];        // vector gather (b64)
    #pragma unroll
    for (int i = 0; i < F / 2; ++i) {
        atomicAdd(&ar[2 * i],     row[i].x * w);
        atomicAdd(&ar[2 * i + 1], row[i].y * w);
    }
}

// ---------------------------------------------------------------------------
// Per-node dense transform with V_WMMA_F32_16X16X4_F32, then L2-norm + ReLU.
//   out[i,:] = normrelu( agg[i,:] @ W_rel^T + b_rel + xin[i,:] @ W_root^T )
// One wave32 owns a 16-node tile. K (Fin = 10 or 20, compile-time) is
// consumed in chunks of 4 (zero padded). 20 output features = two N tiles.
//
// f32 WMMA VGPR layouts (ISA 7.12.2):
//   A 16x4 : lanes 0-15 -> M=lane, regs {K=0,K=1}; lanes 16-31 -> {K=2,K=3}
//   B 4x16 : lanes 0-15 -> N=lane, regs {K=0,K=1}; lanes 16-31 -> {K=2,K=3}
//   D 16x16: reg r, lanes 0-15 -> (M=r, N=lane); lanes 16-31 -> (M=r+8, N=lane-16)
// ---------------------------------------------------------------------------
template<int Fin>
__global__ __launch_bounds__(256)
void lin_norm_wmma(const float* __restrict__ agg, const float* __restrict__ xin,
                   const float* __restrict__ W_rel, const float* __restrict__ b_rel,
                   const float* __restrict__ W_root,
                   float* __restrict__ out, int nNodes) {
    __shared__ __align__(16) float sWrel[FOUT * Fin];
    __shared__ __align__(16) float sWroot[FOUT * Fin];
    __shared__ float sBias[FOUT];
    __shared__ __align__(16) float tileLds[8][16 * FOUT];

    // cooperative stage of weights/bias into LDS
    for (int i = threadIdx.x; i < FOUT * Fin; i += 256) {
        sWrel[i]  = W_rel[i];
        sWroot[i] = W_root[i];
    }
    if (threadIdx.x < FOUT) sBias[threadIdx.x] = b_rel[threadIdx.x];
    __syncthreads();

    const int lane = threadIdx.x & 31;
    const int wave = threadIdx.x >> 5;
    const int tile = blockIdx.x * 8 + wave;   // 16 nodes per tile

    if (tile * 16 < nNodes) {
        const int  m     = lane & 15;          // A row / B col within tile
        const int  khalf = (lane >> 4) * 2;    // lanes 0-15 -> K+0/1, 16-31 -> K+2/3
        const int  node  = tile * 16 + m;
        const int  n1    = 16 + m;             // second N-tile column
        const bool nok   = (n1 < FOUT);        // only m<4 lanes carry tile-1 cols
        const int  n1c   = nok ? n1 : 0;       // clamped (value masked below)

        const v2f* __restrict__ aggRow = (const v2f*)(agg + node * Fin);
        const v2f* __restrict__ xRow   = (const v2f*)(xin + node * Fin);
        const v2f* __restrict__ wr0 = (const v2f*)(sWrel  + m   * Fin);
        const v2f* __restrict__ wo0 = (const v2f*)(sWroot + m   * Fin);
        const v2f* __restrict__ wr1 = (const v2f*)(sWrel  + n1c * Fin);
        const v2f* __restrict__ wo1 = (const v2f*)(sWroot + n1c * Fin);

        v8f c0 = {};   // output features 0..15
        v8f c1 = {};   // output features 16..19 (cols 4..15 stay zero)

        #pragma unroll
        for (int k0 = 0; k0 < Fin; k0 += 4) {
            const int  k  = k0 + khalf;        // even; Fin even => k<Fin ⇔ k+1<Fin
            const bool ok = (k < Fin);         // folds to true for Fin=20
            const int  kh = ok ? (k >> 1) : 0; // clamped v2f index (branch-free)
            const float msk  = ok ? 1.0f : 0.0f;
            const float msk1 = (ok && nok) ? 1.0f : 0.0f;

            v2f aA = aggRow[kh];  aA.x *= msk;  aA.y *= msk;
            v2f aX = xRow[kh];    aX.x *= msk;  aX.y *= msk;
            v2f bR0 = wr0[kh];    bR0.x *= msk;  bR0.y *= msk;
            v2f bO0 = wo0[kh];    bO0.x *= msk;  bO0.y *= msk;
            v2f bR1 = wr1[kh];    bR1.x *= msk1; bR1.y *= msk1;
            v2f bO1 = wo1[kh];    bO1.x *= msk1; bO1.y *= msk1;

            c0 = __builtin_amdgcn_wmma_f32_16x16x4_f32(false, aA, false, bR0, (short)0, c0, false, false);
            c0 = __builtin_amdgcn_wmma_f32_16x16x4_f32(false, aX, false, bO0, (short)0, c0, false, false);
            c1 = __builtin_amdgcn_wmma_f32_16x16x4_f32(false, aA, false, bR1, (short)0, c1, false, false);
            c1 = __builtin_amdgcn_wmma_f32_16x16x4_f32(false, aX, false, bO1, (short)0, c1, false, false);
        }

        // D -> LDS with bias
        const int dm = (lane < 16) ? 0 : 8;
        const int dn = lane & 15;
        #pragma unroll
        for (int r = 0; r < 8; ++r) {
            tileLds[wave][(r + dm) * FOUT + dn] = c0[r] + sBias[dn];
            if (dn < FOUT - 16)
                tileLds[wave][(r + dm) * FOUT + 16 + dn] = c1[r] + sBias[16 + dn];
        }
    }

    __syncthreads();

    // L2-normalize + ReLU: threads 0..127 each own one node of the block's 128
    const int nodeIdx = blockIdx.x * 128 + threadIdx.x;
    if (threadIdx.x < 128 && nodeIdx < nNodes) {
        const int w = threadIdx.x >> 4;
        const int r = threadIdx.x & 15;
        const float* row = &tileLds[w][r * FOUT];
        float ss = 0.0f;
        #pragma unroll
        for (int f = 0; f < FOUT; ++f) ss += row[f] * row[f];
        const float inv = 1.0f / fmaxf(sqrtf(ss), 1e-12f);
        #pragma unroll
        for (int f = 0; f < FOUT; ++f)
            out[(long long)nodeIdx * FOUT + f] = fmaxf(row[f] * inv, 0.0f);
    }
}

// ---------------------------------------------------------------------------
// pooling: per-(node, f) atomics into per-graph sum / max; counts once per node
// emb is post-ReLU (>= 0) so float max == u32 max on raw bits, and the
// zero-initialized max buffer matches the empty-graph -> 0 guard.
// ---------------------------------------------------------------------------
__global__ void pool_kernel(const float* __restrict__ emb, const int* __restrict__ batch,
                            float* __restrict__ sums, float* __restrict__ maxs,
                            float* __restrict__ counts, int nNodes) {
    long long t = (long long)blockIdx.x * blockDim.x + threadIdx.x;
    if (t >= (long long)nNodes * FOUT) return;
    int node = (int)(t / FOUT);
    int f    = (int)(t % FOUT);
    int g    = batch[node];
    float v  = emb[t];
    atomicAdd(&sums[g * FOUT + f], v);
    atomicMax((unsigned int*)&maxs[g * FOUT + f], __float_as_uint(v));
    if (f == 0) atomicAdd(&counts[g], 1.0f);
}

// ---------------------------------------------------------------------------
// final: pooled = [max | mean] (40) ; out[g,:] = pooled @ W_lin^T + b_lin
// ---------------------------------------------------------------------------
__global__ void final_linear(const float* __restrict__ sums, const float* __restrict__ maxs,
                             const float* __restrict__ counts,
                             const float* __restrict__ W_lin, const float* __restrict__ b_lin,
                             float* __restrict__ out) {
    int g = blockIdx.x * blockDim.x + threadIdx.x;
    if (g >= NGRAPHS) return;
    float cinv = 1.0f / fmaxf(counts[g], 1.0f);
    float pooled[40];
    #pragma unroll
    for (int f = 0; f < FOUT; ++f) {
        pooled[f]        = maxs[g * FOUT + f];
        pooled[FOUT + f] = sums[g * FOUT + f] * cinv;
    }
    #pragma unroll
    for (int o = 0; o < 2; ++o) {
        float acc = b_lin[o];
        #pragma unroll
        for (int k = 0; k < 40; ++k) acc += pooled[k] * W_lin[o * 40 + k];
        out[g * 2 + o] = acc;
    }
}

// ---------------------------------------------------------------------------
extern "C" void kernel_launch(void* const* d_in, const int* in_sizes, int n_in,
                              void* d_out, int out_size, void* d_ws, size_t ws_size,
                              hipStream_t stream) {
    const float* x      = (const float*)d_in[0];
    const int*   eidx   = (const int*)d_in[1];
    const int*   batch  = (const int*)d_in[2];
    const float* ew     = (const float*)d_in[3];
    const float* W1_rel = (const float*)d_in[4];
    const float* b1_rel = (const float*)d_in[5];
    const float* W1_root= (const float*)d_in[6];
    const float* W2_rel = (const float*)d_in[7];
    const float* b2_rel = (const float*)d_in[8];
    const float* W2_root= (const float*)d_in[9];
    const float* W3_rel = (const float*)d_in[10];
    const float* b3_rel = (const float*)d_in[11];
    const float* W3_root= (const float*)d_in[12];
    const float* W_lin  = (const float*)d_in[13];
    const float* b_lin  = (const float*)d_in[14];
    float* out = (float*)d_out;

    const int N = in_sizes[2];            // 100000 nodes
    const int E = in_sizes[3];            // 3200000 edges
    const int* src = eidx;                // edge_index[0]
    const int* dst = eidx + E;            // edge_index[1]

    // workspace layout (bytes)
    char* ws = (char*)d_ws;
    float* agg    = (float*)(ws);                          // N*20 f32 = 8 MB
    float* h1     = (float*)(ws + 8 * 1000 * 1000);        // N*20
    float* h2     = (float*)(ws + 16 * 1000 * 1000);       // N*20
    float* emb    = (float*)(ws + 24 * 1000 * 1000);       // N*20
    float* sums   = (float*)(ws + 32 * 1000 * 1000);       // 512*20
    float* maxs   = sums + NGRAPHS * FOUT;
    float* counts = maxs + NGRAPHS * FOUT;

    const int nodeBlocks = (N + 127) / 128;                // 16-node tiles, 8/block
    const int edgeBlocks = (E + 255) / 256;

    // ---- layer 1 (Fin = 10) ----
    zero_f32<<<(N * 10 + 255) / 256, 256, 0, stream>>>(agg, N * 10);
    edge_scatter<10><<<edgeBlocks, 256, 0, stream>>>(src, dst, ew, x, agg, E);
    lin_norm_wmma<10><<<nodeBlocks, 256, 0, stream>>>(agg, x, W1_rel, b1_rel, W1_root, h1, N);

    // ---- layer 2 (Fin = 20) ----
    zero_f32<<<(N * 20 + 255) / 256, 256, 0, stream>>>(agg, N * 20);
    edge_scatter<20><<<edgeBlocks, 256, 0, stream>>>(src, dst, ew, h1, agg, E);
    lin_norm_wmma<20><<<nodeBlocks, 256, 0, stream>>>(agg, h1, W2_rel, b2_rel, W2_root, h2, N);

    // ---- layer 3 (Fin = 20) ----
    zero_f32<<<(N * 20 + 255) / 256, 256, 0, stream>>>(agg, N * 20);
    edge_scatter<20><<<edgeBlocks, 256, 0, stream>>>(src, dst, ew, h2, agg, E);
    lin_norm_wmma<20><<<nodeBlocks, 256, 0, stream>>>(agg, h2, W3_rel, b3_rel, W3_root, emb, N);

    // ---- pooling + final linear ----
    const int poolN = NGRAPHS * FOUT * 2 + NGRAPHS;        // sums+maxs+counts contiguous
    zero_f32<<<(poolN + 255) / 256, 256, 0, stream>>>(sums, poolN);
    long long ptot = (long long)N * FOUT;
    pool_kernel<<<(unsigned)((ptot + 255) / 256), 256, 0, stream>>>(emb, batch, sums, maxs, counts, N);
    final_linear<<<(NGRAPHS + 255) / 256, 256, 0, stream>>>(sums, maxs, counts, W_lin, b_lin, out);
}